// DTCWTInverse_19945828122833
// MI455X (gfx1250) — compile-verified
//
#include <hip/hip_runtime.h>

// ---------------------------------------------------------------------------
// Inverse DTCWT (3 levels) for MI455X / gfx1250, wave32.
// Filtering stages are banded 16x16 GEMM tiles on V_WMMA_F32_16X16X4_F32.
// c2q (complex band -> quad image) is fused into tile gathers; its 1/sqrt(2)
// scale is folded into the banded filter fragments. Symmetric reflection is
// fused into gather indices. Plain-vs-c2q source selection is a template
// parameter so inner loops are branch-free.
//
// Workspace layout (requires >= 3 * 64 MiB):
//   T1 : [0,       64 MiB)   column-pass temp 1
//   T2 : [64 MiB, 128 MiB)   column-pass temp 2
//   LL : [128 MiB,192 MiB)   current lowpass image (ping)
// ---------------------------------------------------------------------------

typedef __attribute__((ext_vector_type(2))) float v2f;
typedef __attribute__((ext_vector_type(8))) float v8f;

#define SQRT_HALF 0.70710678118654752440f

__device__ __forceinline__ int reflect_i(int x, int n) {
  // valid for -n <= x < 2n (pads here are <= 23, n >= 128)
  x = (x < 0) ? (-1 - x) : x;
  x = (x >= n) ? (2 * n - 1 - x) : x;
  return x;
}

// Banded matrix element for colifilt/rowifilt (10-tap pair ha/hb, 4-phase
// 2x-upsampling structure).  m = output index within tile (0..15),
// kk = input index within tile (0..15), relative to x0 = 8*I - 4.
__device__ __forceinline__ float cif_A(int m, int kk,
                                       const float* __restrict__ ha,
                                       const float* __restrict__ hb, int hp) {
  int p = m & 3;
  int q = m >> 2;
  int want_odd = hp ? ((p & 1) ^ 1) : (p & 1);
  int t = ((kk - want_odd) >> 1) - q;
  bool ok = ((kk & 1) == want_odd) && (t >= 0) && (t <= 4);
  int tc = t < 0 ? 0 : (t > 4 ? 4 : t);
  const float* __restrict__ h = ((p == 0) || (p == 2)) ? ha : hb;
  int idx = (p < 2) ? (2 * tc + 1) : (2 * tc);  // hao/hbo vs hae/hbe
  float v = h[idx];
  return ok ? v : 0.0f;
}

// Banded matrix element for colfilter/rowfilter (7- or 5-tap filter h),
// kk relative to x0 = base - 3.
__device__ __forceinline__ float cf_A(int m, int kk,
                                      const float* __restrict__ h, int taps) {
  int t = kk - m - (3 - (taps >> 1));
  bool ok = (t >= 0) && (t < taps);
  int tc = t < 0 ? 0 : (t >= taps ? taps - 1 : t);
  float v = h[tc];
  return ok ? v : 0.0f;
}

// c2q element at reflected quad-row rr for a lane whose quad-column parity is
// cb; p1/p2 are per-lane band pointers already offset to (b, s, 0, j, 0).
// stride = 2*Wb floats per band row.  (1/sqrt2 folded into filter fragments.)
__device__ __forceinline__ float c2q_val(const float* __restrict__ p1,
                                         const float* __restrict__ p2, int rr,
                                         int stride, int cb) {
  int o = (rr >> 1) * stride;
  float w1r = p1[o], w1i = p1[o + 1];
  float w2r = p2[o], w2i = p2[o + 1];
  float even = cb ? (w1i + w2i) : (w1r + w2r);
  float odd = cb ? (w2r - w1r) : (w1i - w2i);
  return (rr & 1) ? odd : even;
}

__device__ __forceinline__ v8f wmma4(v2f a, v2f b, v8f c) {
  // D(16x16,f32) += A(16x4,f32) x B(4x16,f32)
  return __builtin_amdgcn_wmma_f32_16x16x4_f32(false, a, false, b, (short)0, c,
                                               false, false);
}

// ---------------------------------------------------------------------------
// j2plus column pass: out(2r x W) = colifilt(src0, ha0,hb0) +
//                                   colifilt(src1, ha1,hb1, highpass)
// One wave computes one 16x16 output tile: 8 WMMAs (K=16, two sources).
// SRC0_PLAIN selects plain-array vs fused-c2q gathers at compile time.
// ---------------------------------------------------------------------------
template <bool SRC0_PLAIN>
__global__ void __launch_bounds__(256)
j2_col_kernel(const float* __restrict__ src0_plain, const float* __restrict__ yh,
              int s0a, int s0b, int s1a, int s1b, int Hb, int Wb, int r, int W,
              const float* __restrict__ ha0, const float* __restrict__ hb0,
              const float* __restrict__ ha1, const float* __restrict__ hb1,
              float* __restrict__ out, int nTilesR, int nTilesW, int B) {
  int lane = threadIdx.x & 31;
  int wave = threadIdx.x >> 5;
  long tile = (long)blockIdx.x * 8 + wave;
  long total = (long)B * nTilesR * nTilesW;
  if (tile >= total) return;  // wave-uniform: EXEC stays all-ones for WMMA
  int tpi = nTilesR * nTilesW;
  int b = (int)(tile / tpi);
  int t2 = (int)(tile % tpi);
  int I = t2 / nTilesW;
  int Jc = t2 % nTilesW;
  int x0 = 8 * I - 4;  // first input row of this tile (pre-reflection)
  int half = lane >> 4;
  int l16 = lane & 15;
  int koff = half * 2;  // lanes 0-15 hold K 0/1, lanes 16-31 hold K 2/3
  int M = l16;
  int cc = Jc * 16 + l16;

  // Filter fragments; fold c2q's 1/sqrt2 into the fragments of c2q paths.
  const float s0sc = SRC0_PLAIN ? 1.0f : SQRT_HALF;
  v2f a0[4], a1[4];
#pragma unroll
  for (int c = 0; c < 4; ++c) {
    int kk = 4 * c + koff;
    a0[c].x = cif_A(M, kk, ha0, hb0, 0) * s0sc;
    a0[c].y = cif_A(M, kk + 1, ha0, hb0, 0) * s0sc;
    a1[c].x = cif_A(M, kk, ha1, hb1, 1) * SQRT_HALF;
    a1[c].y = cif_A(M, kk + 1, ha1, hb1, 1) * SQRT_HALF;
  }

  // Hoisted per-lane base pointers (only the row index varies per chunk).
  int cb = cc & 1;
  int stride = 2 * Wb;
  int j2 = (cc >> 1) * 2;
  const float* p0 = SRC0_PLAIN ? (src0_plain + (size_t)b * r * W + cc) : nullptr;
  const float* p0a = yh + ((size_t)b * 6 + s0a) * Hb * stride + j2;
  const float* p0b = yh + ((size_t)b * 6 + s0b) * Hb * stride + j2;
  const float* p1a = yh + ((size_t)b * 6 + s1a) * Hb * stride + j2;
  const float* p1b = yh + ((size_t)b * 6 + s1b) * Hb * stride + j2;

  v8f acc = {0.f, 0.f, 0.f, 0.f, 0.f, 0.f, 0.f, 0.f};
#pragma unroll
  for (int c = 0; c < 4; ++c) {
    int kk = 4 * c + koff;
    int r0 = reflect_i(x0 + kk, r);
    int r1 = reflect_i(x0 + kk + 1, r);
    v2f b0, b1;
    if (SRC0_PLAIN) {
      b0.x = p0[(size_t)r0 * W];
      b0.y = p0[(size_t)r1 * W];
    } else {
      b0.x = c2q_val(p0a, p0b, r0, stride, cb);
      b0.y = c2q_val(p0a, p0b, r1, stride, cb);
    }
    b1.x = c2q_val(p1a, p1b, r0, stride, cb);
    b1.y = c2q_val(p1a, p1b, r1, stride, cb);
    acc = wmma4(a0[c], b0, acc);
    acc = wmma4(a1[c], b1, acc);
  }

  int outR = 2 * r;
  float* op = out + ((size_t)b * outR + (size_t)I * 16 + half * 8) * W + cc;
#pragma unroll
  for (int e = 0; e < 8; ++e) op[(size_t)e * W] = acc[e];
}

// ---------------------------------------------------------------------------
// j2plus row pass: out(H x 2Wt) = rowifilt(T1, ha0,hb0) +
//                                 rowifilt(T2, ha1,hb1, highpass)
// Input tile is the A operand, banded filter is the B operand.
// ---------------------------------------------------------------------------
__global__ void __launch_bounds__(256)
j2_row_kernel(const float* __restrict__ T1, const float* __restrict__ T2, int H,
              int Wt, const float* __restrict__ ha0,
              const float* __restrict__ hb0, const float* __restrict__ ha1,
              const float* __restrict__ hb1, float* __restrict__ out,
              int nTilesR, int nTilesW, int B) {
  int lane = threadIdx.x & 31;
  int wave = threadIdx.x >> 5;
  long tile = (long)blockIdx.x * 8 + wave;
  long total = (long)B * nTilesR * nTilesW;
  if (tile >= total) return;
  int tpi = nTilesR * nTilesW;
  int b = (int)(tile / tpi);
  int t2 = (int)(tile % tpi);
  int I = t2 / nTilesW;   // row tile
  int Jc = t2 % nTilesW;  // output-column tile
  int m0 = I * 16;
  int xc0 = 8 * Jc - 4;  // first input column (pre-reflection)
  int half = lane >> 4;
  int l16 = lane & 15;
  int koff = half * 2;

  v2f bf0[4], bf1[4];
#pragma unroll
  for (int c = 0; c < 4; ++c) {
    int kk = 4 * c + koff;
    bf0[c].x = cif_A(l16, kk, ha0, hb0, 0);
    bf0[c].y = cif_A(l16, kk + 1, ha0, hb0, 0);
    bf1[c].x = cif_A(l16, kk, ha1, hb1, 1);
    bf1[c].y = cif_A(l16, kk + 1, ha1, hb1, 1);
  }

  const float* row1 = T1 + ((size_t)b * H + m0 + l16) * Wt;
  const float* row2 = T2 + ((size_t)b * H + m0 + l16) * Wt;
  v8f acc = {0.f, 0.f, 0.f, 0.f, 0.f, 0.f, 0.f, 0.f};
#pragma unroll
  for (int c = 0; c < 4; ++c) {
    int kk = 4 * c + koff;
    int c0 = reflect_i(xc0 + kk, Wt);
    int c1 = reflect_i(xc0 + kk + 1, Wt);
    v2f aA, aB;
    aA.x = row1[c0];
    aA.y = row1[c1];
    aB.x = row2[c0];
    aB.y = row2[c1];
    acc = wmma4(aA, bf0[c], acc);
    acc = wmma4(aB, bf1[c], acc);
  }

  int outW = 2 * Wt;
  float* op = out + ((size_t)b * H + m0 + half * 8) * outW + Jc * 16 + l16;
#pragma unroll
  for (int e = 0; e < 8; ++e) op[(size_t)e * outW] = acc[e];
}

// ---------------------------------------------------------------------------
// j1 column pass: out(r x W) = colfilter(src0, h0[7]) + colfilter(src1, h1[5])
// K = 24 (22 input rows + 2 zero bands) -> 12 WMMAs per tile.
// ---------------------------------------------------------------------------
template <bool SRC0_PLAIN>
__global__ void __launch_bounds__(256)
j1_col_kernel(const float* __restrict__ src0_plain, const float* __restrict__ yh,
              int s0a, int s0b, int s1a, int s1b, int Hb, int Wb, int r, int W,
              const float* __restrict__ h0, int taps0,
              const float* __restrict__ h1, int taps1, float* __restrict__ out,
              int nTilesR, int nTilesW, int B) {
  int lane = threadIdx.x & 31;
  int wave = threadIdx.x >> 5;
  long tile = (long)blockIdx.x * 8 + wave;
  long total = (long)B * nTilesR * nTilesW;
  if (tile >= total) return;
  int tpi = nTilesR * nTilesW;
  int b = (int)(tile / tpi);
  int t2 = (int)(tile % tpi);
  int I = t2 / nTilesW;
  int Jc = t2 % nTilesW;
  int x0 = 16 * I - 3;
  int half = lane >> 4;
  int l16 = lane & 15;
  int koff = half * 2;
  int M = l16;
  int cc = Jc * 16 + l16;

  const float s0sc = SRC0_PLAIN ? 1.0f : SQRT_HALF;
  v2f a0[6], a1[6];
#pragma unroll
  for (int c = 0; c < 6; ++c) {
    int kk = 4 * c + koff;
    a0[c].x = cf_A(M, kk, h0, taps0) * s0sc;
    a0[c].y = cf_A(M, kk + 1, h0, taps0) * s0sc;
    a1[c].x = cf_A(M, kk, h1, taps1) * SQRT_HALF;
    a1[c].y = cf_A(M, kk + 1, h1, taps1) * SQRT_HALF;
  }

  int cb = cc & 1;
  int stride = 2 * Wb;
  int j2 = (cc >> 1) * 2;
  const float* p0 = SRC0_PLAIN ? (src0_plain + (size_t)b * r * W + cc) : nullptr;
  const float* p0a = yh + ((size_t)b * 6 + s0a) * Hb * stride + j2;
  const float* p0b = yh + ((size_t)b * 6 + s0b) * Hb * stride + j2;
  const float* p1a = yh + ((size_t)b * 6 + s1a) * Hb * stride + j2;
  const float* p1b = yh + ((size_t)b * 6 + s1b) * Hb * stride + j2;

  v8f acc = {0.f, 0.f, 0.f, 0.f, 0.f, 0.f, 0.f, 0.f};
#pragma unroll
  for (int c = 0; c < 6; ++c) {
    int kk = 4 * c + koff;
    int r0 = reflect_i(x0 + kk, r);
    int r1 = reflect_i(x0 + kk + 1, r);
    v2f b0, b1;
    if (SRC0_PLAIN) {
      b0.x = p0[(size_t)r0 * W];
      b0.y = p0[(size_t)r1 * W];
    } else {
      b0.x = c2q_val(p0a, p0b, r0, stride, cb);
      b0.y = c2q_val(p0a, p0b, r1, stride, cb);
    }
    b1.x = c2q_val(p1a, p1b, r0, stride, cb);
    b1.y = c2q_val(p1a, p1b, r1, stride, cb);
    acc = wmma4(a0[c], b0, acc);
    acc = wmma4(a1[c], b1, acc);
  }

  float* op = out + ((size_t)b * r + (size_t)I * 16 + half * 8) * W + cc;
#pragma unroll
  for (int e = 0; e < 8; ++e) op[(size_t)e * W] = acc[e];
}

// ---------------------------------------------------------------------------
// j1 row pass: out(H x Wt) = rowfilter(T1, h0[7]) + rowfilter(T2, h1[5])
// ---------------------------------------------------------------------------
__global__ void __launch_bounds__(256)
j1_row_kernel(const float* __restrict__ T1, const float* __restrict__ T2, int H,
              int Wt, const float* __restrict__ h0, int taps0,
              const float* __restrict__ h1, int taps1, float* __restrict__ out,
              int nTilesR, int nTilesW, int B) {
  int lane = threadIdx.x & 31;
  int wave = threadIdx.x >> 5;
  long tile = (long)blockIdx.x * 8 + wave;
  long total = (long)B * nTilesR * nTilesW;
  if (tile >= total) return;
  int tpi = nTilesR * nTilesW;
  int b = (int)(tile / tpi);
  int t2 = (int)(tile % tpi);
  int I = t2 / nTilesW;
  int Jc = t2 % nTilesW;
  int m0 = I * 16;
  int xc0 = 16 * Jc - 3;
  int half = lane >> 4;
  int l16 = lane & 15;
  int koff = half * 2;

  v2f bf0[6], bf1[6];
#pragma unroll
  for (int c = 0; c < 6; ++c) {
    int kk = 4 * c + koff;
    bf0[c].x = cf_A(l16, kk, h0, taps0);
    bf0[c].y = cf_A(l16, kk + 1, h0, taps0);
    bf1[c].x = cf_A(l16, kk, h1, taps1);
    bf1[c].y = cf_A(l16, kk + 1, h1, taps1);
  }

  const float* row1 = T1 + ((size_t)b * H + m0 + l16) * Wt;
  const float* row2 = T2 + ((size_t)b * H + m0 + l16) * Wt;
  v8f acc = {0.f, 0.f, 0.f, 0.f, 0.f, 0.f, 0.f, 0.f};
#pragma unroll
  for (int c = 0; c < 6; ++c) {
    int kk = 4 * c + koff;
    int c0 = reflect_i(xc0 + kk, Wt);
    int c1 = reflect_i(xc0 + kk + 1, Wt);
    v2f aA, aB;
    aA.x = row1[c0];
    aA.y = row1[c1];
    aB.x = row2[c0];
    aB.y = row2[c1];
    acc = wmma4(aA, bf0[c], acc);
    acc = wmma4(aB, bf1[c], acc);
  }

  float* op = out + ((size_t)b * H + m0 + half * 8) * Wt + Jc * 16 + l16;
#pragma unroll
  for (int e = 0; e < 8; ++e) op[(size_t)e * Wt] = acc[e];
}

// ---------------------------------------------------------------------------
extern "C" void kernel_launch(void* const* d_in, const int* in_sizes, int n_in,
                              void* d_out, int out_size, void* d_ws,
                              size_t ws_size, hipStream_t stream) {
  (void)in_sizes;
  (void)n_in;
  (void)out_size;
  (void)ws_size;

  const float* yl = (const float*)d_in[0];
  const float* yh0 = (const float*)d_in[1];
  const float* yh1 = (const float*)d_in[2];
  const float* yh2 = (const float*)d_in[3];
  const float* g0o = (const float*)d_in[4];  // 7 taps
  const float* g1o = (const float*)d_in[5];  // 5 taps
  const float* g0a = (const float*)d_in[6];  // 10 taps
  const float* g0b = (const float*)d_in[7];
  const float* g1a = (const float*)d_in[8];
  const float* g1b = (const float*)d_in[9];

  const int B = 64;                      // N*C
  const size_t IMG = (size_t)512 * 512;  // max per-image footprint (floats)
  float* T1 = (float*)d_ws;
  float* T2 = T1 + (size_t)B * IMG;
  float* LL = T2 + (size_t)B * IMG;
  float* outp = (float*)d_out;

  dim3 blk(256);
  auto nblk = [](long tiles) { return dim3((unsigned)((tiles + 7) / 8)); };

  // ======== Stage 1: _inv_j2plus(yl, yh2) -> LL (256x256) ========
  {
    int r = 128, W = 128, Hb = 64, Wb = 64;
    int ntr = 2 * r / 16, ntw = W / 16;
    long tc = (long)B * ntr * ntw;
    // tmp1 = colifilt(ll, g0b, g0a) + colifilt(lh=c2q(0,5), g1b, g1a, hp)
    j2_col_kernel<true><<<nblk(tc), blk, 0, stream>>>(
        yl, yh2, 0, 0, 0, 5, Hb, Wb, r, W, g0b, g0a, g1b, g1a, T1, ntr, ntw, B);
    // tmp2 = colifilt(hl=c2q(2,3), g0b, g0a) + colifilt(hh=c2q(1,4), ..., hp)
    j2_col_kernel<false><<<nblk(tc), blk, 0, stream>>>(
        nullptr, yh2, 2, 3, 1, 4, Hb, Wb, r, W, g0b, g0a, g1b, g1a, T2, ntr,
        ntw, B);
    int H = 2 * r, Wt = W;
    int rtr = H / 16, rtw = 2 * Wt / 16;
    long tr = (long)B * rtr * rtw;
    j2_row_kernel<<<nblk(tr), blk, 0, stream>>>(T1, T2, H, Wt, g0b, g0a, g1b,
                                                g1a, LL, rtr, rtw, B);
  }

  // ======== Stage 2: _inv_j2plus(LL, yh1) -> LL (512x512) ========
  {
    int r = 256, W = 256, Hb = 128, Wb = 128;
    int ntr = 2 * r / 16, ntw = W / 16;
    long tc = (long)B * ntr * ntw;
    j2_col_kernel<true><<<nblk(tc), blk, 0, stream>>>(
        LL, yh1, 0, 0, 0, 5, Hb, Wb, r, W, g0b, g0a, g1b, g1a, T1, ntr, ntw, B);
    j2_col_kernel<false><<<nblk(tc), blk, 0, stream>>>(
        nullptr, yh1, 2, 3, 1, 4, Hb, Wb, r, W, g0b, g0a, g1b, g1a, T2, ntr,
        ntw, B);
    int H = 2 * r, Wt = W;
    int rtr = H / 16, rtw = 2 * Wt / 16;
    long tr = (long)B * rtr * rtw;
    j2_row_kernel<<<nblk(tr), blk, 0, stream>>>(T1, T2, H, Wt, g0b, g0a, g1b,
                                                g1a, LL, rtr, rtw, B);
  }

  // ======== Stage 3: _inv_j1(LL, yh0) -> d_out (512x512) ========
  {
    int r = 512, W = 512, Hb = 256, Wb = 256;
    int ntr = r / 16, ntw = W / 16;
    long tc = (long)B * ntr * ntw;
    // y1 = colfilter(ll, g0o) + colfilter(lh=c2q(0,5), g1o)
    j1_col_kernel<true><<<nblk(tc), blk, 0, stream>>>(
        LL, yh0, 0, 0, 0, 5, Hb, Wb, r, W, g0o, 7, g1o, 5, T1, ntr, ntw, B);
    // y2 = colfilter(hl=c2q(2,3), g0o) + colfilter(hh=c2q(1,4), g1o)
    j1_col_kernel<false><<<nblk(tc), blk, 0, stream>>>(
        nullptr, yh0, 2, 3, 1, 4, Hb, Wb, r, W, g0o, 7, g1o, 5, T2, ntr, ntw,
        B);
    long tr = (long)B * ntr * ntw;
    j1_row_kernel<<<nblk(tr), blk, 0, stream>>>(T1, T2, r, W, g0o, 7, g1o, 5,
                                                outp, ntr, ntw, B);
  }
}